// SimpleBlock3d_45578192945863
// MI455X (gfx1250) — compile-verified
//
#include <hip/hip_runtime.h>

// ---------------------------------------------------------------------------
// FNO-3D (SimpleBlock3d) for gfx1250: truncated-DFT spectral path + channel
// GEMMs, everything routed through V_WMMA_F32_16X16X4_F32 (fp32 WMMA).
// ---------------------------------------------------------------------------

typedef __attribute__((ext_vector_type(2))) float v2f;
typedef __attribute__((ext_vector_type(8))) float v8f;

#define WMMA_F32(a, b, c) \
  __builtin_amdgcn_wmma_f32_16x16x4_f32(false, (a), false, (b), (short)0, (c), false, false)

#define PI2 6.28318530717958647692f

// Common per-wave fragment coordinates (wave32):
//   lane&15 = lm (A/B/D column-ish index), lane>>4 = half (K / M split)
#define WAVE_COORDS()                          \
  const int lane = threadIdx.x & 31;           \
  const int wave = threadIdx.x >> 5;           \
  const int half = lane >> 4;                  \
  const int lm   = lane & 15;                  \
  const long tile = (long)blockIdx.x * (blockDim.x >> 5) + wave; \
  const long row0 = tile * 16;

// ---------------------------------------------------------------------------
// fc0: rows = (b,x,y,z,n) = 1,048,576 ; K=13 (pad 16) ; N=32
// out h[b][c][sp][n] (sp = x*4096+y*64+z)
// ---------------------------------------------------------------------------
__global__ void k_fc0(const float* __restrict__ x, const float* __restrict__ w,
                      const float* __restrict__ bias, float* __restrict__ h) {
  __shared__ float Bsh[16 * 32];
  for (int t = threadIdx.x; t < 16 * 32; t += blockDim.x) {
    int k = t >> 5, o = t & 31;
    Bsh[t] = (k < 13) ? w[k * 32 + o] : 0.0f;
  }
  __syncthreads();
  WAVE_COORDS();
  const float* arow = x + (row0 + lm) * 13;
  v8f acc0 = {}; v8f acc1 = {};
  for (int kt = 0; kt < 4; ++kt) {
    int k0 = kt * 4 + half * 2;
    v2f a, b0, b1;
    a.x = (k0 < 13) ? arow[k0] : 0.0f;
    a.y = (k0 + 1 < 13) ? arow[k0 + 1] : 0.0f;
    b0.x = Bsh[k0 * 32 + lm];      b0.y = Bsh[(k0 + 1) * 32 + lm];
    b1.x = Bsh[k0 * 32 + 16 + lm]; b1.y = Bsh[(k0 + 1) * 32 + 16 + lm];
    acc0 = WMMA_F32(a, b0, acc0);
    acc1 = WMMA_F32(a, b1, acc1);
  }
  for (int r = 0; r < 8; ++r) {
    long row = row0 + half * 8 + r;
    long b = row >> 19; long rem = row & ((1L << 19) - 1);
    long sp = rem >> 1; int n = (int)(rem & 1);
    h[(b * 32 + lm) * 524288L + sp * 2 + n]      = acc0[r] + bias[lm];
    h[(b * 32 + 16 + lm) * 524288L + sp * 2 + n] = acc1[r] + bias[16 + lm];
  }
}

// ---------------------------------------------------------------------------
// Forward Z: rows = (bc2n, x, y) = 524288 ; K=64 (real z) ; N=16 (8 cplx modes)
// stZ[bc2n][x][y][8k][2] ; B[z][2k+ri] = {cos, -sin}(2*pi*k*z/64)
// ---------------------------------------------------------------------------
__global__ void k_fwdZ(const float* __restrict__ h, float* __restrict__ stZ) {
  __shared__ float Bsh[64 * 16];
  for (int t = threadIdx.x; t < 64 * 16; t += blockDim.x) {
    int z = t >> 4, col = t & 15, k = col >> 1;
    float th = PI2 * (float)(k * z) / 64.0f;
    Bsh[t] = (col & 1) ? -__sinf(th) : __cosf(th);
  }
  __syncthreads();
  WAVE_COORDS();
  long rowA = row0 + lm;
  long bc = rowA >> 12, xy = rowA & 4095;
  const float* arow = h + (bc >> 1) * 524288L + xy * 128 + (bc & 1);
  v8f acc = {};
  for (int kt = 0; kt < 16; ++kt) {
    int k0 = kt * 4 + half * 2;
    v2f a, b;
    a.x = arow[k0 * 2]; a.y = arow[(k0 + 1) * 2];
    b.x = Bsh[k0 * 16 + lm]; b.y = Bsh[(k0 + 1) * 16 + lm];
    acc = WMMA_F32(a, b, acc);
  }
  for (int r = 0; r < 8; ++r)
    stZ[(row0 + half * 8 + r) * 16 + lm] = acc[r];
}

// ---------------------------------------------------------------------------
// Forward Y: rows = (bc2n, x, k) = 65536 ; K=128 (64 cplx y packed) ; N=32
// Packed complex GEMM: Bp[2y][2m]=re, Bp[2y+1][2m]=-im, Bp[2y][2m+1]=im,
// Bp[2y+1][2m+1]=re ; W = exp(-2*pi*i*my*y/64), my in {0..7, 56..63}
// out stY[bc2n][x][16ym][8k][2]
// ---------------------------------------------------------------------------
__global__ void k_fwdY(const float* __restrict__ stZ, float* __restrict__ stY) {
  __shared__ float Bsh[128 * 32];
  for (int t = threadIdx.x; t < 128 * 32; t += blockDim.x) {
    int kr = t >> 5, cc = t & 31;
    int y = kr >> 1, ri = kr & 1, m = cc >> 1, ci = cc & 1;
    int my = (m < 8) ? m : m + 48;
    float th = PI2 * (float)(my * y) / 64.0f;
    float re = __cosf(th), im = -__sinf(th);
    Bsh[t] = (ri == 0) ? ((ci == 0) ? re : im) : ((ci == 0) ? -im : re);
  }
  __syncthreads();
  WAVE_COORDS();
  long rowA = row0 + lm;
  int k = (int)(rowA & 7); long xx = (rowA >> 3) & 63, bc = rowA >> 9;
  const float* ab = stZ + bc * 65536 + xx * 1024 + k * 2;
  v8f acc0 = {}; v8f acc1 = {};
  for (int kt = 0; kt < 32; ++kt) {
    int t0 = kt * 4 + half * 2;      // even
    int yy = t0 >> 1;
    v2f a, b0, b1;
    a.x = ab[yy * 16]; a.y = ab[yy * 16 + 1];
    b0.x = Bsh[t0 * 32 + lm];      b0.y = Bsh[(t0 + 1) * 32 + lm];
    b1.x = Bsh[t0 * 32 + 16 + lm]; b1.y = Bsh[(t0 + 1) * 32 + 16 + lm];
    acc0 = WMMA_F32(a, b0, acc0);
    acc1 = WMMA_F32(a, b1, acc1);
  }
  for (int r = 0; r < 8; ++r) {
    long rr = row0 + half * 8 + r;
    int k2 = (int)(rr & 7); long x2 = (rr >> 3) & 63, bc2 = rr >> 9;
    long ob = bc2 * 16384 + x2 * 256 + k2 * 2;
    int c0 = lm, c1 = 16 + lm;
    stY[ob + (c0 >> 1) * 16 + (c0 & 1)] = acc0[r];
    stY[ob + (c1 >> 1) * 16 + (c1 & 1)] = acc1[r];
  }
}

// ---------------------------------------------------------------------------
// Forward X: rows = (bc2n, ym, k) = 16384 ; K=128 (64 cplx x) ; N=32
// out xf[bc2n][16xm][16ym][8k][2]
// ---------------------------------------------------------------------------
__global__ void k_fwdX(const float* __restrict__ stY, float* __restrict__ xf) {
  __shared__ float Bsh[128 * 32];
  for (int t = threadIdx.x; t < 128 * 32; t += blockDim.x) {
    int kr = t >> 5, cc = t & 31;
    int xx = kr >> 1, ri = kr & 1, m = cc >> 1, ci = cc & 1;
    int mx = (m < 8) ? m : m + 48;
    float th = PI2 * (float)(mx * xx) / 64.0f;
    float re = __cosf(th), im = -__sinf(th);
    Bsh[t] = (ri == 0) ? ((ci == 0) ? re : im) : ((ci == 0) ? -im : re);
  }
  __syncthreads();
  WAVE_COORDS();
  long rowA = row0 + lm;
  int k = (int)(rowA & 7); int ym = (int)((rowA >> 3) & 15); long bc = rowA >> 7;
  const float* ab = stY + bc * 16384 + ym * 16 + k * 2;
  v8f acc0 = {}; v8f acc1 = {};
  for (int kt = 0; kt < 32; ++kt) {
    int t0 = kt * 4 + half * 2;
    int xx = t0 >> 1;
    v2f a, b0, b1;
    a.x = ab[xx * 256]; a.y = ab[xx * 256 + 1];
    b0.x = Bsh[t0 * 32 + lm];      b0.y = Bsh[(t0 + 1) * 32 + lm];
    b1.x = Bsh[t0 * 32 + 16 + lm]; b1.y = Bsh[(t0 + 1) * 32 + 16 + lm];
    acc0 = WMMA_F32(a, b0, acc0);
    acc1 = WMMA_F32(a, b1, acc1);
  }
  for (int r = 0; r < 8; ++r) {
    long rr = row0 + half * 8 + r;
    int k2 = (int)(rr & 7); int ym2 = (int)((rr >> 3) & 15); long bc2 = rr >> 7;
    long ob = bc2 * 4096 + ym2 * 16 + k2 * 2;
    int c0 = lm, c1 = 16 + lm;
    xf[ob + (c0 >> 1) * 256 + (c0 & 1)] = acc0[r];
    xf[ob + (c1 >> 1) * 256 + (c1 & 1)] = acc1[r];
  }
}

// ---------------------------------------------------------------------------
// Spectral multiply: per-mode 32x32 complex channel contraction.
// gid = (((b*2+n)*16+xm)*16+ym)*8+k ; quadrant q = (ym>=8)*2 + (xm>=8)
// sw float layout: (((((q*32+i)*32+o)*512 + (m1*8+m2)*8+m3))*2+n)*2 + ri
// ---------------------------------------------------------------------------
__global__ void k_specmul(const float* __restrict__ xf, const float* __restrict__ sw,
                          float* __restrict__ yf) {
  __shared__ float av[64];
  int gid = blockIdx.x;
  int k  = gid & 7;         int ym = (gid >> 3) & 15;
  int xm = (gid >> 7) & 15; int n  = (gid >> 11) & 1; int b = (gid >> 12) & 1;
  int t = threadIdx.x;
  if (t < 64) {
    int i = t >> 1, ri = t & 1;
    long bc = (long)(b * 32 + i) * 2 + n;
    av[t] = xf[bc * 4096 + xm * 256 + ym * 16 + k * 2 + ri];
  }
  __syncthreads();
  if (t < 32) {
    int o = t;
    int q = ((ym >= 8) ? 2 : 0) + ((xm >= 8) ? 1 : 0);
    long inner = (long)((xm & 7) * 8 + (ym & 7)) * 8 + k;
    float accr = 0.0f, acci = 0.0f;
    for (int i = 0; i < 32; ++i) {
      long widx = (((long)(q * 32 + i) * 32 + o) * 512 + inner) * 4 + n * 2;
      float wr = sw[widx], wi = sw[widx + 1];
      float xr = av[2 * i], xi = av[2 * i + 1];
      accr += xr * wr - xi * wi;
      acci += xr * wi + xi * wr;
    }
    long bo = (long)(b * 32 + o) * 2 + n;
    yf[bo * 4096 + xm * 256 + ym * 16 + k * 2 + 0] = accr;
    yf[bo * 4096 + xm * 256 + ym * 16 + k * 2 + 1] = acci;
  }
}

// ---------------------------------------------------------------------------
// Inverse X: rows = (bc2n, ym, k) = 16384 ; K=32 (16 cplx modes) ; N=128 (64 cplx x)
// W = exp(+2*pi*i*mx*x/64) ; out stIX[bc2n][x][16ym][8k][2]
// ---------------------------------------------------------------------------
__global__ void k_invX(const float* __restrict__ yf, float* __restrict__ stIX) {
  __shared__ float Bsh[32 * 128];
  for (int t = threadIdx.x; t < 32 * 128; t += blockDim.x) {
    int kr = t >> 7, cc = t & 127;
    int m = kr >> 1, rk = kr & 1, xx = cc >> 1, rc = cc & 1;
    int mx = (m < 8) ? m : m + 48;
    float th = PI2 * (float)(mx * xx) / 64.0f;
    float re = __cosf(th), im = __sinf(th);
    Bsh[t] = (rk == 0) ? ((rc == 0) ? re : im) : ((rc == 0) ? -im : re);
  }
  __syncthreads();
  WAVE_COORDS();
  long rowA = row0 + lm;
  int k = (int)(rowA & 7); int ym = (int)((rowA >> 3) & 15); long bc = rowA >> 7;
  const float* ab = yf + bc * 4096 + ym * 16 + k * 2;
  v8f acc[8] = {};
  for (int kt = 0; kt < 8; ++kt) {
    int t0 = kt * 4 + half * 2;
    int xm = t0 >> 1;
    v2f a; a.x = ab[xm * 256]; a.y = ab[xm * 256 + 1];
    for (int nt = 0; nt < 8; ++nt) {
      v2f b; b.x = Bsh[t0 * 128 + nt * 16 + lm]; b.y = Bsh[(t0 + 1) * 128 + nt * 16 + lm];
      acc[nt] = WMMA_F32(a, b, acc[nt]);
    }
  }
  for (int r = 0; r < 8; ++r) {
    long rr = row0 + half * 8 + r;
    int k2 = (int)(rr & 7); int ym2 = (int)((rr >> 3) & 15); long bc2 = rr >> 7;
    long ob = bc2 * 16384 + ym2 * 16 + k2 * 2;
    for (int nt = 0; nt < 8; ++nt) {
      int cc = nt * 16 + lm;
      stIX[ob + (long)(cc >> 1) * 256 + (cc & 1)] = acc[nt][r];
    }
  }
}

// ---------------------------------------------------------------------------
// Inverse Y: rows = (bc2n, x, k) = 65536 ; K=32 ; N=128 (64 cplx y)
// out stIY[bc2n][x][y][8k][2]
// ---------------------------------------------------------------------------
__global__ void k_invY(const float* __restrict__ stIX, float* __restrict__ stIY) {
  __shared__ float Bsh[32 * 128];
  for (int t = threadIdx.x; t < 32 * 128; t += blockDim.x) {
    int kr = t >> 7, cc = t & 127;
    int m = kr >> 1, rk = kr & 1, yy = cc >> 1, rc = cc & 1;
    int my = (m < 8) ? m : m + 48;
    float th = PI2 * (float)(my * yy) / 64.0f;
    float re = __cosf(th), im = __sinf(th);
    Bsh[t] = (rk == 0) ? ((rc == 0) ? re : im) : ((rc == 0) ? -im : re);
  }
  __syncthreads();
  WAVE_COORDS();
  long rowA = row0 + lm;
  int k = (int)(rowA & 7); long xx = (rowA >> 3) & 63, bc = rowA >> 9;
  const float* ab = stIX + bc * 16384 + xx * 256 + k * 2;
  v8f acc[8] = {};
  for (int kt = 0; kt < 8; ++kt) {
    int t0 = kt * 4 + half * 2;
    int ym = t0 >> 1;
    v2f a; a.x = ab[ym * 16]; a.y = ab[ym * 16 + 1];
    for (int nt = 0; nt < 8; ++nt) {
      v2f b; b.x = Bsh[t0 * 128 + nt * 16 + lm]; b.y = Bsh[(t0 + 1) * 128 + nt * 16 + lm];
      acc[nt] = WMMA_F32(a, b, acc[nt]);
    }
  }
  for (int r = 0; r < 8; ++r) {
    long rr = row0 + half * 8 + r;
    int k2 = (int)(rr & 7); long x2 = (rr >> 3) & 63, bc2 = rr >> 9;
    long ob = bc2 * 65536 + x2 * 1024 + k2 * 2;
    for (int nt = 0; nt < 8; ++nt) {
      int cc = nt * 16 + lm;
      stIY[ob + (cc >> 1) * 16 + (cc & 1)] = acc[nt][r];
    }
  }
}

// ---------------------------------------------------------------------------
// Inverse Z (irfft, Hermitian doubling + 1/(64^3) scale): rows = (bc2n,x,y)
// K=16 (8 cplx modes packed) ; N=64 (real z) ; out xp[b][c][sp][n]
// ---------------------------------------------------------------------------
__global__ void k_invZ(const float* __restrict__ stIY, float* __restrict__ xp) {
  __shared__ float Bsh[16 * 64];
  for (int t = threadIdx.x; t < 16 * 64; t += blockDim.x) {
    int kr = t >> 6, z = t & 63;
    int k = kr >> 1, ri = kr & 1;
    float s = ((k == 0) ? 1.0f : 2.0f) * (1.0f / 262144.0f);
    float th = PI2 * (float)(k * z) / 64.0f;
    Bsh[t] = (ri == 0) ? s * __cosf(th) : -s * __sinf(th);
  }
  __syncthreads();
  WAVE_COORDS();
  long rowA = row0 + lm;
  long bc = rowA >> 12, xy = rowA & 4095;
  const float* ab = stIY + bc * 65536 + xy * 16;
  v8f acc[4] = {};
  for (int kt = 0; kt < 4; ++kt) {
    int t0 = kt * 4 + half * 2;
    v2f a; a.x = ab[t0]; a.y = ab[t0 + 1];
    for (int nt = 0; nt < 4; ++nt) {
      v2f b; b.x = Bsh[t0 * 64 + nt * 16 + lm]; b.y = Bsh[(t0 + 1) * 64 + nt * 16 + lm];
      acc[nt] = WMMA_F32(a, b, acc[nt]);
    }
  }
  for (int r = 0; r < 8; ++r) {
    long rr = row0 + half * 8 + r;
    long bc2 = rr >> 12, xy2 = rr & 4095;
    long ob = (bc2 >> 1) * 524288 + xy2 * 128 + (bc2 & 1);
    for (int nt = 0; nt < 4; ++nt)
      xp[ob + (nt * 16 + lm) * 2] = acc[nt][r];
  }
}

// ---------------------------------------------------------------------------
// conv1x1 + bias + per-(n,channel) sum/sumsq partials (LDS -> global atomics)
// rows = (b,x,y,z,n) = 1,048,576 ; K=32 ; N=32
// ---------------------------------------------------------------------------
__global__ void k_conv_stats(const float* __restrict__ h, const float* __restrict__ w,
                             const float* __restrict__ bias, float* __restrict__ y,
                             float* __restrict__ stats) {
  __shared__ float Bsh[32 * 32];
  __shared__ float ls[128];  // [n][o][sum,sumsq]
  for (int t = threadIdx.x; t < 1024; t += blockDim.x)
    Bsh[t] = w[(t & 31) * 32 + (t >> 5)];  // transpose: Bsh[c*32+o] = w[o][c]
  for (int t = threadIdx.x; t < 128; t += blockDim.x) ls[t] = 0.0f;
  __syncthreads();
  WAVE_COORDS();
  long rowA = row0 + lm;
  long b = rowA >> 19; long rem = rowA & ((1L << 19) - 1);
  long sp = rem >> 1; int n = (int)(rem & 1);
  const float* ab = h + b * 16777216L + sp * 2 + n;
  v8f acc0 = {}; v8f acc1 = {};
  for (int kt = 0; kt < 8; ++kt) {
    int k0 = kt * 4 + half * 2;
    v2f a, b0, b1;
    a.x = ab[(long)k0 * 524288];
    a.y = ab[(long)(k0 + 1) * 524288];
    b0.x = Bsh[k0 * 32 + lm];      b0.y = Bsh[(k0 + 1) * 32 + lm];
    b1.x = Bsh[k0 * 32 + 16 + lm]; b1.y = Bsh[(k0 + 1) * 32 + 16 + lm];
    acc0 = WMMA_F32(a, b0, acc0);
    acc1 = WMMA_F32(a, b1, acc1);
  }
  for (int r = 0; r < 8; ++r) {
    long rr = row0 + half * 8 + r;
    long b2 = rr >> 19; long rem2 = rr & ((1L << 19) - 1);
    long sp2 = rem2 >> 1; int n2 = (int)(rem2 & 1);
    float v0 = acc0[r] + bias[lm];
    float v1 = acc1[r] + bias[16 + lm];
    y[(b2 * 32 + lm) * 524288L + sp2 * 2 + n2]      = v0;
    y[(b2 * 32 + 16 + lm) * 524288L + sp2 * 2 + n2] = v1;
    atomicAdd(&ls[(n2 * 32 + lm) * 2 + 0], v0);
    atomicAdd(&ls[(n2 * 32 + lm) * 2 + 1], v0 * v0);
    atomicAdd(&ls[(n2 * 32 + 16 + lm) * 2 + 0], v1);
    atomicAdd(&ls[(n2 * 32 + 16 + lm) * 2 + 1], v1 * v1);
  }
  __syncthreads();
  for (int t = threadIdx.x; t < 128; t += blockDim.x) atomicAdd(&stats[t], ls[t]);
}

__global__ void k_bnfin(const float* __restrict__ stats, float* __restrict__ bn) {
  int t = threadIdx.x;
  if (t < 64) {
    float s = stats[t * 2], ss = stats[t * 2 + 1];
    float mu = s * (1.0f / 524288.0f);
    float var = ss * (1.0f / 524288.0f) - mu * mu;
    bn[t * 2] = mu;
    bn[t * 2 + 1] = rsqrtf(var + 1e-5f);
  }
}

// out = relu(xp + g*(y-mu)*invstd + beta), in place on y (becomes next h)
__global__ void k_combine(const float* __restrict__ xp, float* __restrict__ y,
                          const float* __restrict__ bn, const float* __restrict__ g,
                          const float* __restrict__ beta) {
  long i = (long)blockIdx.x * blockDim.x + threadIdx.x;
  int n = (int)(i & 1);
  int c = (int)((i >> 19) & 31);
  float mu = bn[(n * 32 + c) * 2], is = bn[(n * 32 + c) * 2 + 1];
  float v = y[i];
  float r = xp[i] + g[c] * ((v - mu) * is) + beta[c];
  y[i] = (r > 0.0f) ? r : 0.0f;
}

// ---------------------------------------------------------------------------
// fc1 (32->128, WMMA) + relu + fc2 (128->1, cross-lane dot) fused
// ---------------------------------------------------------------------------
__global__ void k_fc12(const float* __restrict__ h, const float* __restrict__ w1,
                       const float* __restrict__ b1, const float* __restrict__ w2,
                       const float* __restrict__ b2v, float* __restrict__ out) {
  __shared__ float Bsh[32 * 128];
  __shared__ float W2[128];
  for (int t = threadIdx.x; t < 4096; t += blockDim.x) Bsh[t] = w1[t];  // [c][o]
  for (int t = threadIdx.x; t < 128; t += blockDim.x) W2[t] = w2[t];
  __syncthreads();
  WAVE_COORDS();
  long rowA = row0 + lm;
  long b = rowA >> 19; long rem = rowA & ((1L << 19) - 1);
  long sp = rem >> 1; int n = (int)(rem & 1);
  const float* ab = h + b * 16777216L + sp * 2 + n;
  v8f acc[8] = {};
  for (int kt = 0; kt < 8; ++kt) {
    int k0 = kt * 4 + half * 2;
    v2f a;
    a.x = ab[(long)k0 * 524288];
    a.y = ab[(long)(k0 + 1) * 524288];
    for (int nt = 0; nt < 8; ++nt) {
      v2f bb; bb.x = Bsh[k0 * 128 + nt * 16 + lm]; bb.y = Bsh[(k0 + 1) * 128 + nt * 16 + lm];
      acc[nt] = WMMA_F32(a, bb, acc[nt]);
    }
  }
  float bias2 = b2v[0];
  for (int r = 0; r < 8; ++r) {
    float p = 0.0f;
    for (int nt = 0; nt < 8; ++nt) {
      int col = nt * 16 + lm;
      float v = acc[nt][r] + b1[col];
      v = (v > 0.0f) ? v : 0.0f;
      p += v * W2[col];
    }
    for (int off = 8; off; off >>= 1) p += __shfl_xor(p, off, 16);
    if (lm == 0) out[row0 + half * 8 + r] = p + bias2;
  }
}

// ---------------------------------------------------------------------------
extern "C" void kernel_launch(void* const* d_in, const int* in_sizes, int n_in,
                              void* d_out, int out_size, void* d_ws, size_t ws_size,
                              hipStream_t stream) {
  const float* x     = (const float*)d_in[0];
  const float* fc0_w = (const float*)d_in[1];
  const float* fc0_b = (const float*)d_in[2];
  const float* sw[3] = {(const float*)d_in[3], (const float*)d_in[4], (const float*)d_in[5]};
  const float* cw[3] = {(const float*)d_in[6], (const float*)d_in[8], (const float*)d_in[10]};
  const float* cb[3] = {(const float*)d_in[7], (const float*)d_in[9], (const float*)d_in[11]};
  const float* g[3]  = {(const float*)d_in[12], (const float*)d_in[14], (const float*)d_in[16]};
  const float* bt[3] = {(const float*)d_in[13], (const float*)d_in[15], (const float*)d_in[17]};
  const float* fc1_w = (const float*)d_in[18];
  const float* fc1_b = (const float*)d_in[19];
  const float* fc2_w = (const float*)d_in[20];
  const float* fc2_b = (const float*)d_in[21];
  float* outp = (float*)d_out;

  float* W = (float*)d_ws;
  float* bigA = W;                      // 33,554,432 floats (h ping)
  float* bigB = bigA + 33554432L;       // 33,554,432 (xp)
  float* bigC = bigB + 33554432L;       // 33,554,432 (y / h pong)
  float* stZ  = bigC + 33554432L;       //  8,388,608 (fwd Z out / inv Y out)
  float* stY  = stZ + 8388608L;         //  2,097,152 (fwd Y out / inv X out)
  float* xfB  = stY + 2097152L;         //    524,288
  float* yfB  = xfB + 524288L;          //    524,288
  float* stats = yfB + 524288L;         //        128
  float* bn    = stats + 128;           //        128

  k_fc0<<<16384, 128, 0, stream>>>(x, fc0_w, fc0_b, bigA);

  float* hin = bigA;
  float* ybuf = bigC;
  float* other = bigC;  // tracks free big buffer for next y
  for (int blk = 0; blk < 3; ++blk) {
    k_fwdZ<<<8192, 128, 0, stream>>>(hin, stZ);
    k_fwdY<<<1024, 128, 0, stream>>>(stZ, stY);
    k_fwdX<<<256, 128, 0, stream>>>(stY, xfB);
    k_specmul<<<8192, 64, 0, stream>>>(xfB, sw[blk], yfB);
    k_invX<<<256, 128, 0, stream>>>(yfB, stY);
    k_invY<<<1024, 128, 0, stream>>>(stY, stZ);
    k_invZ<<<8192, 128, 0, stream>>>(stZ, bigB);
    hipMemsetAsync(stats, 0, 128 * sizeof(float), stream);
    k_conv_stats<<<16384, 128, 0, stream>>>(hin, cw[blk], cb[blk], ybuf, stats);
    k_bnfin<<<1, 64, 0, stream>>>(stats, bn);
    k_combine<<<131072, 256, 0, stream>>>(bigB, ybuf, bn, g[blk], bt[blk]);
    // result now lives in ybuf; old hin becomes the next y buffer
    other = hin;
    hin = ybuf;
    ybuf = other;
  }
  // after 3 blocks: hin = bigC (blk0: A->C, blk1: C->A, blk2: A->C)
  k_fc12<<<16384, 128, 0, stream>>>(hin, fc1_w, fc1_b, fc2_w, fc2_b, outp);
}